// BackProjection_58025008169123
// MI455X (gfx1250) — compile-verified
//
#include <hip/hip_runtime.h>

#ifndef __has_builtin
#define __has_builtin(x) 0
#endif

#if __has_builtin(__builtin_amdgcn_global_load_async_to_lds_b32) && \
    __has_builtin(__builtin_amdgcn_s_wait_asynccnt)
#define HAVE_ASYNC_LDS 1
#else
#define HAVE_ASYNC_LDS 0
#endif

#define G       128
#define NLORS   16384
#define LPB     2                 // LORs per block
#define THREADS (LPB * G)         // 256 threads = 8 wave32 waves

// 0.5 / SIGMA^2 with SIGMA^2 = 9*pi/4  ->  2/(9*pi)
#define INV2SIG2 0.070735530f
#define VPIX     1.5625f          // 200/128
#define ORIG     (-100.0f)
#define INV_VPIX 0.64f            // 1/1.5625

#if HAVE_ASYNC_LDS
typedef __attribute__((address_space(1))) int gas_int;  // global addrspace
typedef __attribute__((address_space(3))) int las_int;  // LDS addrspace
#endif

// C0,C1,C2: column permutation mapping rotated coords -> original lor columns.
// S0,S1,SK: output flat strides for rotated indices (i0,i1,k), already fused
// with the reference's final transpose, so we scatter straight into d_out.
template <int C0, int C1, int C2, int S0, int S1, int SK>
__global__ __launch_bounds__(THREADS)
void bp_kernel(const float* __restrict__ lors,
               const float* __restrict__ proj,
               float* __restrict__ out)
{
    __shared__ float sh[LPB * 8];

    const int tid = threadIdx.x;
    const int k   = tid & (G - 1);   // plane index along dominant axis
    const int l   = tid >> 7;        // which LOR of this block

    // ---- Stage 2 LORs (6 endpoint floats + 1 proj each) into LDS ----
    if (tid < LPB * 8) {
        const int li = tid >> 3;
        const int c  = tid & 7;
        if (c < 7) {
            const float* g = (c < 6)
                ? (lors + (size_t)(blockIdx.x * LPB + li) * 6 + c)
                : (proj + (size_t)(blockIdx.x * LPB + li));
#if HAVE_ASYNC_LDS
            float* gnc = const_cast<float*>(g);
            __builtin_amdgcn_global_load_async_to_lds_b32(
                (gas_int*)gnc,
                (las_int*)&sh[li * 8 + c],
                /*offset=*/0, /*cpol=*/0);
#else
            sh[li * 8 + c] = *g;
#endif
        }
    }
#if HAVE_ASYNC_LDS
    __builtin_amdgcn_s_wait_asynccnt(0);
#endif
    __syncthreads();

    const float* s = &sh[l * 8];
    const float p1x = s[C0],     p1y = s[C1],     p1z = s[C2];
    const float p2x = s[3 + C0], p2y = s[3 + C1], p2z = s[3 + C2];
    const float pr  = s[6];

    float dz = p2z - p1z;
    if (fabsf(dz) < 1e-6f) dz = 1e-6f;            // matches jnp.where(|dz|<1e-6, 1e-6, dz)

    const float zc = ORIG + ((float)k + 0.5f) * VPIX;
    const float t  = (zc - p1z) / dz;
    if (!(t >= 0.0f && t <= 1.0f)) return;        // valid_t

    const float cx = p1x + t * (p2x - p1x);
    const float cy = p1y + t * (p2y - p1y);

    const int ix0 = (int)rintf((cx - ORIG) * INV_VPIX - 0.5f);  // round-half-even
    const int iy0 = (int)rintf((cy - ORIG) * INV_VPIX - 0.5f);

    // Separable Gaussian: 6 v_exp_f32 instead of 9 exps
    float wx[3], wy[3];
#pragma unroll
    for (int d = 0; d < 3; ++d) {
        const float xcv = ORIG + ((float)(ix0 + d - 1) + 0.5f) * VPIX;
        const float ycv = ORIG + ((float)(iy0 + d - 1) + 0.5f) * VPIX;
        const float dx = xcv - cx;
        const float dy = ycv - cy;
        wx[d] = __expf(-(dx * dx) * INV2SIG2);
        wy[d] = __expf(-(dy * dy) * INV2SIG2);
    }

    // Fire-and-forget global_atomic_add_f32 (no return -> STOREcnt only)
#pragma unroll
    for (int di = 0; di < 3; ++di) {
        const int ix = ix0 + di - 1;
        if ((unsigned)ix >= (unsigned)G) continue;
        const float wxp = wx[di] * pr;
#pragma unroll
        for (int dj = 0; dj < 3; ++dj) {
            const int iy = iy0 + dj - 1;
            if ((unsigned)iy >= (unsigned)G) continue;
            atomicAdd(out + (size_t)ix * S0 + (size_t)iy * S1 + (size_t)k * SK,
                      wxp * wy[dj]);
        }
    }
}

extern "C" void kernel_launch(void* const* d_in, const int* in_sizes, int n_in,
                              void* d_out, int out_size, void* d_ws, size_t ws_size,
                              hipStream_t stream)
{
    // setup_inputs order: image, xlors, ylors, zlors, xproj, yproj, zproj
    // (image is unused by the reference)
    const float* xlors = (const float*)d_in[1];
    const float* ylors = (const float*)d_in[2];
    const float* zlors = (const float*)d_in[3];
    const float* xproj = (const float*)d_in[4];
    const float* yproj = (const float*)d_in[5];
    const float* zproj = (const float*)d_in[6];
    float* out = (float*)d_out;

    // d_out is poisoned (0xAA) and NOT re-poisoned between replays: zero it
    // every call so the scatter-accumulate is deterministic.
    (void)hipMemsetAsync(d_out, 0, (size_t)out_size * sizeof(float), stream);

    dim3 grid(NLORS / LPB), block(THREADS);

    // axis x: rotated cols (y,z,x) = (1,2,0); bp[i0,i1,k] -> out_x[i1,k,i0]
    bp_kernel<1, 2, 0, 1,   G*G, G><<<grid, block, 0, stream>>>(xlors, xproj, out);
    // axis y: rotated cols (x,z,y) = (0,2,1); bp[i0,i1,k] -> out_y[i1,i0,k]
    bp_kernel<0, 2, 1, G,   G*G, 1><<<grid, block, 0, stream>>>(ylors, yproj, out + (size_t)G*G*G);
    // axis z: identity;                     bp[i0,i1,k] -> out_z[i0,i1,k]
    bp_kernel<0, 1, 2, G*G, G,   1><<<grid, block, 0, stream>>>(zlors, zproj, out + 2*(size_t)G*G*G);
}